// Solver_23493471109757
// MI455X (gfx1250) — compile-verified
//
#include <hip/hip_runtime.h>
#include <math.h>

// ---------------------------------------------------------------------------
// Problem constants (fixed by the reference)
// ---------------------------------------------------------------------------
#define NEDGE           17      // 5 + 5 + 3 + 4 obstacle edges
#define SAMPLE_NUM      30
#define SEGS_PER_WAVE   8       // 8 segs * 30 samples = 240 = 15 * 16 (exact WMMA tiles)
#define TILES_PER_WAVE  15
#define WAVES_PER_BLOCK 8
#define BLOCK_THREADS   256
#define SEGS_PER_BLOCK  (SEGS_PER_WAVE * WAVES_PER_BLOCK)   // 64

typedef __attribute__((ext_vector_type(2))) float v2f;
typedef __attribute__((ext_vector_type(8))) float v8f;

struct EdgeConsts {
    float dix[NEDGE], diy[NEDGE], bias[NEDGE];   // fxy: outward normal + (0.1 - q.d)
    float qx[NEDGE],  qy[NEDGE];                 // wzy: edge start
    float sx[NEDGE],  sy[NEDGE];                 // wzy: edge direction
};

__device__ __forceinline__ float wave32_sum(float v) {
    v += __shfl_xor(v, 16, 32);
    v += __shfl_xor(v,  8, 32);
    v += __shfl_xor(v,  4, 32);
    v += __shfl_xor(v,  2, 32);
    v += __shfl_xor(v,  1, 32);
    return v;
}

// ---------------------------------------------------------------------------
// Main kernel: one wave handles 8 segments.
//  fxy -> 15x V_WMMA_F32_16X16X4_F32 (edges x samples affine transform)
//  wzy -> 4 lanes per segment (edges strided by 4)
//  lp  -> 1 lane per segment
// Per-block partials: partials[3*blockIdx + {0,1,2}] = {wzy, fxy, lp}
// ---------------------------------------------------------------------------
__global__ __launch_bounds__(BLOCK_THREADS)
void Solver_path_kernel(const float* __restrict__ path,
                        const float* __restrict__ start_pt,
                        const float* __restrict__ end_pt,
                        EdgeConsts ec,
                        float* __restrict__ partials,
                        int nseg)
{
    __shared__ float  s_dix[16], s_diy[16], s_bias[16];   // A-matrix build
    __shared__ float4 s_wedge[NEDGE];                     // {qx, qy, sx, sy}
    __shared__ float2 s_pts[WAVES_PER_BLOCK][SEGS_PER_WAVE + 1];
    __shared__ float  s_wsum[WAVES_PER_BLOCK][3];

    const int tid  = threadIdx.x;
    const int wave = tid >> 5;
    const int lane = tid & 31;
    const int col  = lane & 15;

    // Stage edge constants.
    if (tid < 16) {
        s_dix[tid]  = ec.dix[tid];
        s_diy[tid]  = ec.diy[tid];
        s_bias[tid] = ec.bias[tid];
    }
    if (tid < NEDGE)
        s_wedge[tid] = make_float4(ec.qx[tid], ec.qy[tid], ec.sx[tid], ec.sy[tid]);

    const int s0 = blockIdx.x * SEGS_PER_BLOCK + wave * SEGS_PER_WAVE;

    // Stage the 9 fullpath points this wave needs (segments s0..s0+7).
    if (lane <= SEGS_PER_WAVE) {
        int i = s0 + lane;                 // fullpath index, 0..nseg
        float2 pt;
        if (i == 0)         pt = make_float2(start_pt[0], start_pt[1]);
        else if (i == nseg) pt = make_float2(end_pt[0],  end_pt[1]);
        else                pt = make_float2(path[2*(i-1)], path[2*(i-1)+1]);
        s_pts[wave][lane] = pt;
    }
    __syncthreads();

    // A matrix (16 edges x K=4 = [dix, diy, bias, 0]); fp32 16x4 layout:
    // lanes 0-15 carry K0,K1  |  lanes 16-31 carry K2,K3 for row M = lane%16.
    v2f A;
    if (lane < 16) { A.x = s_dix[lane];        A.y = s_diy[lane]; }
    else           { A.x = s_bias[lane - 16];  A.y = 0.0f;        }

    const float dix16  = ec.dix[16];
    const float diy16  = ec.diy[16];
    const float bias16 = ec.bias[16];
    const bool  lowHalf = (lane < 16);

    float acc_fxy = 0.0f;

    // Incremental (segment-offset, sample-index) tracking: local = 16*t + col.
    int sOff = 0;
    int k    = col;                          // col <= 15 < 30

    #pragma unroll
    for (int t = 0; t < TILES_PER_WAVE; ++t) {
        float2 p  = s_pts[wave][sOff];
        float2 pn = s_pts[wave][sOff + 1];
        float  rx = pn.x - p.x, ry = pn.y - p.y;

        // fracs = [1, 1/30, 2/30, ..., 29/30]
        float frac = (k == 0) ? 1.0f : (float)k * (1.0f / (float)SAMPLE_NUM);
        float sxx = fmaf(frac, rx, p.x);
        float syy = fmaf(frac, ry, p.y);

        // B matrix (K=4 x 16 samples = [x, y, 1, 0] per column), branchless.
        v2f B;
        B.x = lowHalf ? sxx : 1.0f;
        B.y = lowHalf ? syy : 0.0f;

        v8f c = {0.f, 0.f, 0.f, 0.f, 0.f, 0.f, 0.f, 0.f};
        c = __builtin_amdgcn_wmma_f32_16x16x4_f32(
                /*neg_a=*/false, A, /*neg_b=*/false, B,
                /*c_mod=*/(short)0, c, /*reuse_a=*/false, /*reuse_b=*/false);

        // D[edge][sample]: lane j -> edges 0..7, lane j+16 -> edges 8..15.
        float e0 = fmaxf(c[0], 0.f), e1 = fmaxf(c[1], 0.f);
        float e2 = fmaxf(c[2], 0.f), e3 = fmaxf(c[3], 0.f);
        float e4 = fmaxf(c[4], 0.f), e5 = fmaxf(c[5], 0.f);
        float e6 = fmaxf(c[6], 0.f), e7 = fmaxf(c[7], 0.f);

        // Shared partial products, fully branchless epilogue.
        float t01  = e0 * e1;              // low: ob0 head | up: ob1 edges 8-9
        float t234 = e2 * e3 * e4;         // low: ob0 tail | up: ob2 (10-12)
        float tail = e5 * e6 * e7;         // low: ob1 (5-7) | up: ob3 (13-15)
        float e16v = fmaxf(fmaf(sxx, dix16, fmaf(syy, diy16, bias16)), 0.0f);

        float other = __shfl_xor(t01, 16, 32);   // lower receives upper's e0*e1
        float contrib = lowHalf ? fmaf(tail, other, t01 * t234)   // ob0 + ob1
                                : fmaf(tail * e16v, 1.0f, t234);  // ob3 + ob2
        acc_fxy += contrib;

        // Advance (sOff, k) by 16 samples, branchless.
        k += 16;
        int wrap = (k >= SAMPLE_NUM);
        k    = wrap ? k - SAMPLE_NUM : k;
        sOff = wrap ? sOff + 1 : sOff;
    }

    // ------------------------------------------------------------------
    // wzy + lpath: 4 lanes per segment; edges sub, sub+4, sub+8, sub+12,
    // plus edge 16 on sub==0 lanes.
    // ------------------------------------------------------------------
    const int segOff = lane >> 2;          // 0..7
    const int sub    = lane & 3;

    float2 p  = s_pts[wave][segOff];
    float2 pn = s_pts[wave][segOff + 1];
    float  rx = pn.x - p.x, ry = pn.y - p.y;

    float acc_lp  = (sub == 0) ? sqrtf(1e-6f + rx * rx + ry * ry) : 0.0f;
    float acc_wzy = 0.0f;

    #pragma unroll
    for (int i = 0; i < 5; ++i) {
        int e = sub + 4 * i;               // 0..16; i==4 only valid for sub==0
        if (i == 4 && sub != 0) break;
        float4 w = s_wedge[e];             // {qx, qy, sx, sy}
        float denom = rx * w.w - ry * w.z + 1e-8f;
        float qpx = w.x - p.x;
        float qpy = w.y - p.y;
        float tt = (qpx * w.w - qpy * w.z) / denom;
        float uu = (qpx * ry  - qpy * rx ) / denom;
        float st_ = (tt / fabsf(tt)) * ((tt - 1.0f) / fabsf(tt - 1.0f));
        float su_ = (uu / fabsf(uu)) * ((uu - 1.0f) / fabsf(uu - 1.0f));
        float collt = 1.0f - (st_ + 1.0f) * 0.5f;
        float collu = 1.0f - (su_ + 1.0f) * 0.5f;
        acc_wzy += collt * collu;
    }

    // ------------------------------------------------------------------
    // Deterministic reduction: wave32 shuffle butterfly -> LDS -> wave 0.
    // ------------------------------------------------------------------
    float w_wzy = wave32_sum(acc_wzy);
    float w_fxy = wave32_sum(acc_fxy);
    float w_lp  = wave32_sum(acc_lp);
    if (lane == 0) {
        s_wsum[wave][0] = w_wzy;
        s_wsum[wave][1] = w_fxy;
        s_wsum[wave][2] = w_lp;
    }
    __syncthreads();

    if (tid < WAVES_PER_BLOCK) {
        float a = s_wsum[tid][0];
        float b = s_wsum[tid][1];
        float d = s_wsum[tid][2];
        a += __shfl_xor(a, 4, 32); b += __shfl_xor(b, 4, 32); d += __shfl_xor(d, 4, 32);
        a += __shfl_xor(a, 2, 32); b += __shfl_xor(b, 2, 32); d += __shfl_xor(d, 2, 32);
        a += __shfl_xor(a, 1, 32); b += __shfl_xor(b, 1, 32); d += __shfl_xor(d, 1, 32);
        if (tid == 0) {
            partials[3 * blockIdx.x + 0] = a;
            partials[3 * blockIdx.x + 1] = b;
            partials[3 * blockIdx.x + 2] = d;
        }
    }
}

// ---------------------------------------------------------------------------
// Final deterministic reduce: one block, fixed accumulation order.
// ---------------------------------------------------------------------------
__global__ __launch_bounds__(BLOCK_THREADS)
void Solver_reduce_kernel(const float* __restrict__ partials,
                          float* __restrict__ out, int nblocks)
{
    __shared__ float sh[BLOCK_THREADS];
    const int tid = threadIdx.x;
    for (int comp = 0; comp < 3; ++comp) {
        float a = 0.0f;
        for (int j = tid; j < nblocks; j += BLOCK_THREADS)
            a += partials[3 * j + comp];
        sh[tid] = a; __syncthreads();
        for (int s = BLOCK_THREADS / 2; s > 0; s >>= 1) {
            if (tid < s) sh[tid] += sh[tid + s];
            __syncthreads();
        }
        if (tid == 0) out[comp] = sh[0];
        __syncthreads();
    }
}

// ---------------------------------------------------------------------------
// Host launcher
// ---------------------------------------------------------------------------
extern "C" void kernel_launch(void* const* d_in, const int* in_sizes, int n_in,
                              void* d_out, int out_size, void* d_ws, size_t ws_size,
                              hipStream_t stream)
{
    (void)n_in; (void)out_size; (void)ws_size;
    const float* path  = (const float*)d_in[0];
    const float* start = (const float*)d_in[1];
    const float* endp  = (const float*)d_in[2];
    float*       out   = (float*)d_out;

    const int npath = in_sizes[0] / 2;   // 262143
    const int nseg  = npath + 1;         // 262144 (multiple of 64)

    // Precompute obstacle edge constants on host (deterministic, capture-safe).
    static const int   obN = 4;
    static const int   obCnt[4] = {5, 5, 3, 4};
    static const float obPts[NEDGE][2] = {
        {1.23f, 3.47f}, {1.75f, 4.00f}, {2.10f, 3.63f}, {1.58f, 2.30f}, {1.40f, 2.67f},
        {4.65f, 5.98f}, {4.00f, 6.48f}, {4.52f, 7.68f}, {5.06f, 7.73f}, {5.90f, 6.95f},
        {6.78f, 3.40f}, {7.78f, 5.10f}, {7.78f, 3.76f},
        {4.00f, 3.00f}, {4.35f, 3.35f}, {4.80f, 3.45f}, {4.37f, 2.75f}
    };

    EdgeConsts ec;
    int base = 0, e = 0;
    for (int o = 0; o < obN; ++o) {
        for (int i = 0; i < obCnt[o]; ++i) {
            int j = (i + 1) % obCnt[o];
            float qx = obPts[base + i][0], qy = obPts[base + i][1];
            float nx = obPts[base + j][0], ny = obPts[base + j][1];
            float sx = nx - qx, sy = ny - qy;
            float dx = sy, dy = -sx;                       // outward normal dir
            float inv = 1.0f / sqrtf(dx * dx + dy * dy);
            dx *= inv; dy *= inv;
            ec.dix[e] = dx; ec.diy[e] = dy;
            ec.bias[e] = 0.1f - (qx * dx + qy * dy);       // fold -q.d + 0.1
            ec.qx[e] = qx;  ec.qy[e] = qy;
            ec.sx[e] = sx;  ec.sy[e] = sy;
            ++e;
        }
        base += obCnt[o];
    }

    const int nblocks = nseg / SEGS_PER_BLOCK;             // 4096
    float* partials = (float*)d_ws;                        // 3 * nblocks floats

    Solver_path_kernel<<<nblocks, BLOCK_THREADS, 0, stream>>>(
        path, start, endp, ec, partials, nseg);
    Solver_reduce_kernel<<<1, BLOCK_THREADS, 0, stream>>>(partials, out, nblocks);
}